// GroupNorm_19490561589617
// MI455X (gfx1250) — compile-verified
//
#include <hip/hip_runtime.h>

typedef __attribute__((ext_vector_type(2))) float v2f;
typedef __attribute__((ext_vector_type(4))) float f4;
typedef __attribute__((ext_vector_type(8))) float v8f;

#define GROUPS 8
#define CPG 8
#define HW 16384              // 128*128 spatial samples per (b,c)
#define PLANE_FLOATS 32768    // HW * D, D=2 (contiguous plane per (b,c))
#define NCH 64
#define NB 16
#define NPLANES (NB * NCH)    // 1024
#define NGRP_TOT (NB * GROUPS)// 128
#define EPS 1e-5f

// ---------------------------------------------------------------------------
// Wave32 all-lane sum using V_WMMA_F32_16X16X4_F32.
// A (16x4 f32) layout (ISA 7.12.2): lane L<16 holds row L; VGPR0 = K0 (lanes
// 0-15) / K2 (lanes 16-31); VGPR1 = K1 / K3.  Setting both A VGPRs to the
// per-lane partial p makes row L sum to 2*(p_L + p_{L+16}); B == 0.5 in every
// slot (layout-insensitive since all-constant) cancels the factor 2.
// D layout: VGPR r, lanes 0-15 -> row r (= p_r + p_{r+16}), lanes 16-31 ->
// row r+8.  Summing the 8 D VGPRs gives the half-wave sums; one xor-16
// shuffle completes the 32-lane total, broadcast to all lanes.
// Requires EXEC == all ones (callers are fully converged).
// ---------------------------------------------------------------------------
__device__ __forceinline__ float wave_reduce32(float p) {
  v2f a; a[0] = p;    a[1] = p;
  v2f b; b[0] = 0.5f; b[1] = 0.5f;
  v8f c = {};
  v8f d = __builtin_amdgcn_wmma_f32_16x16x4_f32(
      /*neg_a=*/false, a, /*neg_b=*/false, b,
      /*c_mod=*/(short)0, c, /*reuse_a=*/false, /*reuse_b=*/false);
  float e = d[0] + d[1] + d[2] + d[3] + d[4] + d[5] + d[6] + d[7];
  e += __shfl_xor(e, 16, 32);
  return e;
}

// ---------------------------------------------------------------------------
// Pass 1: per (b,c) plane raw moments: Sx, Sy, Qxx, Qxy, Qyy.
// One block per plane; f4 (b128) loads; default temporal hint keeps x in L2
// (x = 128 MiB fits the 192 MiB L2) for re-use by pass 3.
// ---------------------------------------------------------------------------
__global__ __launch_bounds__(256) void gn_stats(const float* __restrict__ x,
                                                float* __restrict__ stats) {
  const int bc = blockIdx.x;
  const f4* __restrict__ p = (const f4*)(x + (size_t)bc * PLANE_FLOATS);

  float sx = 0.f, sy = 0.f, qxx = 0.f, qxy = 0.f, qyy = 0.f;
#pragma unroll 4
  for (int i = threadIdx.x; i < PLANE_FLOATS / 4; i += 256) {
    f4 v = p[i];                 // {x0,y0,x1,y1}: two vec2 samples
    sx  += v.x + v.z;
    sy  += v.y + v.w;
    qxx += v.x * v.x + v.z * v.z;
    qxy += v.x * v.y + v.z * v.w;
    qyy += v.y * v.y + v.w * v.w;
  }

  // Wave-level reduction on the matrix pipe (EXEC all ones here).
  float r0 = wave_reduce32(sx);
  float r1 = wave_reduce32(sy);
  float r2 = wave_reduce32(qxx);
  float r3 = wave_reduce32(qxy);
  float r4 = wave_reduce32(qyy);

  __shared__ float sm[8][5];
  const int wid = threadIdx.x >> 5;
  const int lane = threadIdx.x & 31;
  if (lane == 0) {
    sm[wid][0] = r0; sm[wid][1] = r1; sm[wid][2] = r2;
    sm[wid][3] = r3; sm[wid][4] = r4;
  }
  __syncthreads();
  if (threadIdx.x < 5) {
    float t = 0.f;
#pragma unroll
    for (int w = 0; w < 8; ++w) t += sm[w][threadIdx.x];
    stats[bc * 5 + threadIdx.x] = t;
  }
}

// ---------------------------------------------------------------------------
// Pass 2: per (b,g) closed-form 2x2 inverse-sqrt of the covariance.
// cov = Q/n - mu mu^T ; W = V diag((lambda+eps)^-1/2) V^T = alpha*I + beta*cov
// with beta = (s2-s1)/(l2-l1) (derivative limit when degenerate).
// ---------------------------------------------------------------------------
__global__ void gn_params(const float* __restrict__ stats,
                          float* __restrict__ prm) {
  const int t = threadIdx.x;
  if (t >= NGRP_TOT) return;
  const int b = t >> 3, g = t & 7;

  const float* __restrict__ s = stats + (size_t)(b * NCH + g * CPG) * 5;
  float Sx = 0.f, Sy = 0.f, Qxx = 0.f, Qxy = 0.f, Qyy = 0.f;
#pragma unroll
  for (int c = 0; c < CPG; ++c) {
    Sx  += s[c * 5 + 0];
    Sy  += s[c * 5 + 1];
    Qxx += s[c * 5 + 2];
    Qxy += s[c * 5 + 3];
    Qyy += s[c * 5 + 4];
  }
  const float inv_n = 1.0f / (float)(CPG * HW);   // 1/131072
  const float mux = Sx * inv_n, muy = Sy * inv_n;
  const float a  = Qxx * inv_n - mux * mux;
  const float bb = Qxy * inv_n - mux * muy;
  const float cc = Qyy * inv_n - muy * muy;

  const float tr2   = 0.5f * (a + cc);
  const float delta = sqrtf(fmaxf(0.f, 0.25f * (a - cc) * (a - cc) + bb * bb));
  const float l1 = tr2 - delta, l2 = tr2 + delta;
  const float s1 = 1.0f / sqrtf(l1 + EPS);
  const float s2 = 1.0f / sqrtf(l2 + EPS);

  float beta;
  if (delta > 1e-12f) {
    beta = (s2 - s1) / (2.0f * delta);
  } else {                       // degenerate eigenvalues: f'(lambda)
    const float le = tr2 + EPS;
    beta = -0.5f / (le * sqrtf(le));
  }
  const float alpha = s1 - beta * l1;

  prm[t * 5 + 0] = alpha + beta * a;   // w00
  prm[t * 5 + 1] = beta * bb;          // w01 (= w10)
  prm[t * 5 + 2] = alpha + beta * cc;  // w11
  prm[t * 5 + 3] = mux;
  prm[t * 5 + 4] = muy;
}

// ---------------------------------------------------------------------------
// Pass 3: out = scale[c] * W (x - mu_g) + bias[c] * mean_vec[b,c]
// Folded to 2 FMAs per output component.  x re-read hits L2 (resident from
// pass 1); output uses non-temporal stores so it doesn't evict x.
// ---------------------------------------------------------------------------
__global__ __launch_bounds__(256) void gn_apply(
    const float* __restrict__ x, const float* __restrict__ scale,
    const float* __restrict__ bias, const float* __restrict__ stats,
    const float* __restrict__ prm, float* __restrict__ out) {
  const int bc = blockIdx.x;
  const int b = bc >> 6, c = bc & 63;
  const int g = c >> 3;

  const float* __restrict__ w = prm + (size_t)(b * GROUPS + g) * 5;
  const float w00 = w[0], w01 = w[1], w11 = w[2];
  const float mux = w[3], muy = w[4];
  const float sc = scale[c], bi = bias[c];
  const float mvx = stats[bc * 5 + 0] * (1.0f / (float)HW);
  const float mvy = stats[bc * 5 + 1] * (1.0f / (float)HW);

  const float a00 = sc * w00, a01 = sc * w01, a11 = sc * w11;
  const float cx = bi * mvx - (a00 * mux + a01 * muy);
  const float cy = bi * mvy - (a01 * mux + a11 * muy);

  const f4* __restrict__ xp = (const f4*)(x + (size_t)bc * PLANE_FLOATS);
  f4* __restrict__ op = (f4*)(out + (size_t)bc * PLANE_FLOATS);

#pragma unroll 4
  for (int i = threadIdx.x; i < PLANE_FLOATS / 4; i += 256) {
    f4 v = xp[i];
    f4 o;
    o.x = a00 * v.x + a01 * v.y + cx;
    o.y = a01 * v.x + a11 * v.y + cy;
    o.z = a00 * v.z + a01 * v.w + cx;
    o.w = a01 * v.z + a11 * v.w + cy;
    __builtin_nontemporal_store(o, op + i);   // TH=NT: keep x resident in L2
  }
}

// ---------------------------------------------------------------------------
extern "C" void kernel_launch(void* const* d_in, const int* in_sizes, int n_in,
                              void* d_out, int out_size, void* d_ws,
                              size_t ws_size, hipStream_t stream) {
  const float* x     = (const float*)d_in[0];
  const float* scale = (const float*)d_in[1];
  const float* bias  = (const float*)d_in[2];
  float* out = (float*)d_out;

  float* stats = (float*)d_ws;              // NPLANES*5 = 5120 floats
  float* prm   = stats + (size_t)NPLANES * 5; // NGRP_TOT*5 = 640 floats

  gn_stats <<<NPLANES, 256, 0, stream>>>(x, stats);
  gn_params<<<1, 128, 0, stream>>>(stats, prm);
  gn_apply <<<NPLANES, 256, 0, stream>>>(x, scale, bias, stats, prm, out);
}